// NegativeSampler_60636348285273
// MI455X (gfx1250) — compile-verified
//
#include <hip/hip_runtime.h>
#include <stdint.h>

#define WAVES_PER_BLOCK 8
#define ROW_FLOATS_MAX 256   // D = 256 in the reference

// ---------------------------------------------------------------------------
// Exact replication of jax.random.uniform(jax.random.key(42), (B, M)) bits:
// threefry2x32 with key (0, 42), 20 rounds.
// ---------------------------------------------------------------------------
__device__ __forceinline__ void threefry2x32_key42(unsigned c0, unsigned c1,
                                                   unsigned& o0, unsigned& o1) {
  const unsigned ks0 = 0u;
  const unsigned ks1 = 42u;
  const unsigned ks2 = 0x1BD11BDAu ^ ks0 ^ ks1;
  const unsigned ks[3] = {ks0, ks1, ks2};
  const int rotA[4] = {13, 15, 26, 6};
  const int rotB[4] = {17, 29, 16, 24};
  unsigned x0 = c0 + ks0;
  unsigned x1 = c1 + ks1;
#pragma unroll
  for (int g = 0; g < 5; ++g) {
    const int* rot = (g & 1) ? rotB : rotA;
#pragma unroll
    for (int rr = 0; rr < 4; ++rr) {
      x0 += x1;
      x1 = (x1 << rot[rr]) | (x1 >> (32 - rot[rr]));
      x1 ^= x0;
    }
    x0 += ks[(g + 1) % 3];
    x1 += ks[(g + 2) % 3] + (unsigned)(g + 1);
  }
  o0 = x0;
  o1 = x1;
}

// ---------------------------------------------------------------------------
// Kernel A: for each label value l in [0,32), build the stable list of row
// indices k with label[k] != l (original order), and its count.
// One block per label value; wave32 ballot + cross-wave prefix in LDS.
// ---------------------------------------------------------------------------
__global__ __launch_bounds__(1024) void build_neg_lists(
    const int* __restrict__ label, int* __restrict__ lists,
    int* __restrict__ negcnt, int Bn) {
  const int l = blockIdx.x;
  const int k = threadIdx.x;
  __shared__ int wcnt[32];

  const bool valid = (k < Bn);
  const int myLab = valid ? label[k] : l;       // pad lanes count as "positive"
  const bool isNeg = valid && (myLab != l);

  const unsigned m32 = (unsigned)__ballot(isNeg);   // wave32: low 32 bits valid
  const int wave = k >> 5;
  const int lane = k & 31;
  if (lane == 0) wcnt[wave] = __popc(m32);
  __syncthreads();

  int base = 0;
  for (int w = 0; w < wave; ++w) base += wcnt[w];
  if (isNeg) {
    const int pos = base + __popc(m32 & ((1u << lane) - 1u));
    lists[l * Bn + pos] = k;
  }
  if (k == 0) {
    const int nW = (blockDim.x + 31) >> 5;
    int tot = 0;
    for (int w = 0; w < nW; ++w) tot += wcnt[w];
    negcnt[l] = tot;
  }
}

// ---------------------------------------------------------------------------
// Kernel B: 2D grid — blockIdx.y = anchor i, blockIdx.x tiles j.
// One wave32 per output row [i, j]: resolve the gathered source row (stable
// j-th negative, or threefry-resampled tail), then DMA the 1KB row through
// LDS with CDNA5 async global<->LDS instructions (ASYNCcnt), NT stores.
// ---------------------------------------------------------------------------
__global__ __launch_bounds__(256) void neg_gather(
    const float* __restrict__ embs, const int* __restrict__ label,
    const int* __restrict__ lists, const int* __restrict__ negcnt,
    float* __restrict__ out, int Bn, int Dn, int maxc,
    unsigned half, unsigned n) {
  __shared__ __align__(16) float lds[WAVES_PER_BLOCK * ROW_FLOATS_MAX];

  const int wslot = threadIdx.x >> 5;
  const int lane = threadIdx.x & 31;
  const int i = blockIdx.y;                             // anchor row
  const int j = blockIdx.x * WAVES_PER_BLOCK + wslot;   // negative slot
  if (j >= maxc) return;                                // wave-uniform

  const int li = label[i];
  const int neg = negcnt[li];

  int idx = j;
  if (j >= neg) {
    // flat position p in the (B, max_count) uniform array
    const unsigned p = (unsigned)i * (unsigned)maxc + (unsigned)j;
    unsigned c0, c1;
    int which;
    if (p < half) {
      c0 = p;
      const unsigned t = p + half;
      c1 = (t < n) ? t : 0u;   // odd-size zero pad, as in JAX
      which = 0;
    } else {
      c0 = p - half;
      c1 = p;
      which = 1;
    }
    unsigned o0, o1;
    threefry2x32_key42(c0, c1, o0, o1);
    const unsigned bits = which ? o1 : o0;
    const float u = __uint_as_float((bits >> 9) | 0x3f800000u) - 1.0f;
    int rj = (int)(u * (float)neg);   // f32 mul + trunc, matches astype(int32)
    if (rj > neg - 1) rj = neg - 1;
    idx = rj;
  }
  const int srcRow = lists[li * Bn + idx];

  const size_t r = (size_t)i * (size_t)maxc + (size_t)j;  // output row id
  const unsigned ldsBase =
      (unsigned)(uintptr_t)(&lds[wslot * ROW_FLOATS_MAX]);
  const unsigned la = ldsBase + (unsigned)(lane * 16);
  const int nbytes = Dn * 4;

  if (nbytes == 1024) {
    // Fast path (D=256): one address pair, INST_OFFSET applies to both the
    // LDS and the global address of async ops (ISA 15.18), so two issues
    // move the full 1KB row per wave.
    unsigned long long ga =
        (unsigned long long)(uintptr_t)(embs + (size_t)srcRow * Dn) +
        (unsigned)(lane * 16);
    asm volatile(
        "global_load_async_to_lds_b128 %0, %1, off\n\t"
        "global_load_async_to_lds_b128 %0, %1, off offset:512"
        :: "v"(la), "v"(ga)
        : "memory");
#if __has_builtin(__builtin_amdgcn_s_wait_asynccnt)
    __builtin_amdgcn_s_wait_asynccnt(0);
#else
    asm volatile("s_wait_asynccnt 0" ::: "memory");
#endif
    unsigned long long go =
        (unsigned long long)(uintptr_t)(out + r * Dn) +
        (unsigned)(lane * 16);
    asm volatile(
        "global_store_async_from_lds_b128 %0, %1, off th:TH_STORE_NT\n\t"
        "global_store_async_from_lds_b128 %0, %1, off offset:512 th:TH_STORE_NT"
        :: "v"(go), "v"(la)
        : "memory");
  } else {
    // Generic fallback for other row sizes
    const char* src = (const char*)(embs + (size_t)srcRow * Dn);
    char* dst = (char*)(out + r * Dn);
    for (int ofs = lane * 16; ofs < nbytes; ofs += 512) {
      unsigned lo = ldsBase + (unsigned)ofs;
      unsigned long long ga = (unsigned long long)(uintptr_t)(src + ofs);
      asm volatile("global_load_async_to_lds_b128 %0, %1, off"
                   :: "v"(lo), "v"(ga)
                   : "memory");
    }
#if __has_builtin(__builtin_amdgcn_s_wait_asynccnt)
    __builtin_amdgcn_s_wait_asynccnt(0);
#else
    asm volatile("s_wait_asynccnt 0" ::: "memory");
#endif
    for (int ofs = lane * 16; ofs < nbytes; ofs += 512) {
      unsigned lo = ldsBase + (unsigned)ofs;
      unsigned long long ga = (unsigned long long)(uintptr_t)(dst + ofs);
      asm volatile("global_store_async_from_lds_b128 %0, %1, off th:TH_STORE_NT"
                   :: "v"(ga), "v"(lo)
                   : "memory");
    }
  }
  // No trailing wait needed: S_ENDPGM performs an implicit wait-idle.
}

// ---------------------------------------------------------------------------
// Launcher
// ---------------------------------------------------------------------------
extern "C" void kernel_launch(void* const* d_in, const int* in_sizes, int n_in,
                              void* d_out, int out_size, void* d_ws, size_t ws_size,
                              hipStream_t stream) {
  const float* embs = (const float*)d_in[0];
  const int* label = (const int*)d_in[1];

  const int Bn = in_sizes[1];               // 1024
  const int Dn = in_sizes[0] / Bn;          // 256
  const int maxc = out_size / in_sizes[0];  // max_count = out / (B*D)

  int* lists = (int*)d_ws;                  // [32][Bn] negative index lists
  int* negcnt = lists + (size_t)32 * Bn;    // [32] counts

  int bthreads = (Bn + 31) & ~31;
  if (bthreads > 1024) bthreads = 1024;
  hipLaunchKernelGGL(build_neg_lists, dim3(32), dim3(bthreads), 0, stream,
                     label, lists, negcnt, Bn);

  const unsigned n = (unsigned)Bn * (unsigned)maxc;
  const unsigned half = (n + (n & 1u)) >> 1;
  const int blocks_x = (maxc + WAVES_PER_BLOCK - 1) / WAVES_PER_BLOCK;
  hipLaunchKernelGGL(neg_gather, dim3(blocks_x, Bn), dim3(256), 0, stream,
                     embs, label, lists, negcnt, (float*)d_out,
                     Bn, Dn, maxc, half, n);
}